// ChamferDistLoss_66022237274638
// MI455X (gfx1250) — compile-verified
//
#include <hip/hip_runtime.h>
#include <math.h>

// CDNA5 / gfx1250, wave32. Chamfer loss via V_WMMA_F32_16X16X4_F32.
// d^2 = |a|^2 + |b|^2 - 2 a.b ; |b|^2 folded into the K=4 slot of the WMMA,
// |a|^2 added after the (monotone) min, sqrt applied only to final row-mins.

// Relax FP strictness: avoids v_max_num_f32 canonicalize ops around fminf and
// re-enables v_min3_num_f32 fusion of the min-accumulation chains.
#pragma float_control(precise, off)

typedef __attribute__((ext_vector_type(2))) float v2f;
typedef __attribute__((ext_vector_type(4))) float v4f;
typedef __attribute__((ext_vector_type(8))) float v8f;

#define BATCH   4
#define NPTS    4096
#define TILE    16
#define NTILES  (NPTS / TILE)              // 256
#define NPAIRS  4
#define ROWS    2                          // row tiles per wave (B-load reuse)
#define NBLK_PK ((NPAIRS * BATCH * NPTS) / 256)        // 256 blocks
#define NBLK_MN (NPAIRS * BATCH * (NTILES / ROWS))     // 2048 waves
#define NBLK_FB (NPAIRS * BATCH * NTILES)              // 4096 (fallback)
#define PACK_V2F ((size_t)NPAIRS * BATCH * NTILES * 32)  // 131072 v2f = 1 MB

// ---------------------------------------------------------------------------
// Pre-pack each point set into the WMMA B-operand lane layout, grouped as
// tile-PAIRS so the main loop loads two tiles with one b128:
//   v4f element (set,b,J,lane): .xy = operand of tile 2J,  lane l
//                               .zw = operand of tile 2J+1, lane l
// where per tile: lane l<16 : (-2x, -2y) of point tile*16+l
//                 lane l>=16: (-2z, |b|^2) of point tile*16+(l-16)
// ---------------------------------------------------------------------------
__global__ __launch_bounds__(256)
void pack_b_operands(const float* __restrict__ s0, const float* __restrict__ s1,
                     const float* __restrict__ s2, const float* __restrict__ s3,
                     v2f* __restrict__ packed)
{
    const int tid = blockIdx.x * 256 + threadIdx.x;   // 65536 threads
    const int set = tid >> 14;
    const int b   = (tid >> 12) & (BATCH - 1);
    const int p   = tid & (NPTS - 1);

    const float* src;
    switch (set) {
        case 0:  src = s0; break;
        case 1:  src = s1; break;
        case 2:  src = s2; break;
        default: src = s3; break;
    }
    const float* q = src + ((size_t)b * NPTS + p) * 3;
    const float x = q[0], y = q[1], z = q[2];
    const float b2 = x * x + y * y + z * z;

    const int j    = p >> 4;        // column tile
    const int J    = j >> 1;        // tile pair
    const int half = j & 1;         // which half of the v4f
    const int c    = p & 15;        // column within tile

    // v2f-granular index: ((group * NTILES/2 + J) * 32 + lane) * 2 + half
    const size_t base = ((size_t)(set * BATCH + b) * (NTILES / 2) + J) * 64;
    v2f lo;  lo.x  = -2.0f * x; lo.y  = -2.0f * y;
    v2f hiv; hiv.x = -2.0f * z; hiv.y = b2;
    packed[base + (size_t)c * 2 + half]        = lo;
    packed[base + (size_t)(16 + c) * 2 + half] = hiv;
}

// ---------------------------------------------------------------------------
// Main pass: one wave per (pair, batch, pair-of-row-tiles). Steady state per
// tile-pair: 1x global_load_b128 + 4x WMMA + 16x v_min3.
// ---------------------------------------------------------------------------
__global__ __launch_bounds__(32)
void chamfer_rowmin_wmma_packed(const float* __restrict__ pt_src,
                                const float* __restrict__ pt_ref,
                                const v2f*   __restrict__ packed,
                                float*       __restrict__ partials)
{
    const int lin  = blockIdx.x;
    const int g    = lin & (NTILES / ROWS - 1);     // row-tile group (0..127)
    const int b    = (lin >> 7) & (BATCH - 1);
    const int pair = lin >> 9;

    // B set per pair: 0->pt_ref(1), 1->pt_src(0), 2->points_src(2), 3->points_ref(3)
    const int bset = (pair < 2) ? (pair ^ 1) : pair;
    const float* A = ((pair & 1) ? pt_ref : pt_src) + (size_t)b * NPTS * 3;
    const v4f*  Bt = (const v4f*)packed
                   + (size_t)(bset * BATCH + b) * (NTILES / 2) * 32;

    const int  lane = threadIdx.x;
    const bool hi   = lane >= 16;
    const int  l16  = lane & 15;

    // A operands (loop invariant): A row = [x, y, z, 1]
    v2f   Aop[ROWS];
    float a2[ROWS];
    #pragma unroll
    for (int t = 0; t < ROWS; ++t) {
        const int row = (g * ROWS + t) * TILE + l16;
        const float* ap = A + (size_t)row * 3;
        const float ax = ap[0], ay = ap[1], az = ap[2];
        a2[t] = ax * ax + ay * ay + az * az;
        Aop[t].x = hi ? az   : ax;
        Aop[t].y = hi ? 1.0f : ay;
    }

    v8f runmin[ROWS];
    #pragma unroll
    for (int t = 0; t < ROWS; ++t)
        #pragma unroll
        for (int r = 0; r < 8; ++r) runmin[t][r] = 3.0e37f;

    #pragma unroll 2
    for (int J = 0; J < NTILES / 2; ++J) {
        const v4f q = Bt[J * 32 + lane];            // b128: two tiles' operands
        v2f b0; b0.x = q.x; b0.y = q.y;             // tile 2J
        v2f b1; b1.x = q.z; b1.y = q.w;             // tile 2J+1
        #pragma unroll
        for (int t = 0; t < ROWS; ++t) {
            v8f c0 = {}, c1 = {};
            v8f d0 = __builtin_amdgcn_wmma_f32_16x16x4_f32(
                false, Aop[t], false, b0, (short)0, c0, false, false);
            v8f d1 = __builtin_amdgcn_wmma_f32_16x16x4_f32(
                false, Aop[t], false, b1, (short)0, c1, false, false);
            #pragma unroll
            for (int r = 0; r < 8; ++r)             // fuses to v_min3_num_f32
                runmin[t][r] = fminf(runmin[t][r], fminf(d0[r], d1[r]));
        }
    }

    // Finalize each row tile: min across the 16 lanes of each half, add the
    // deferred |a|^2, sqrt, and sum the 16 row distances.
    float tot = 0.0f;
    #pragma unroll
    for (int t = 0; t < ROWS; ++t) {
        #pragma unroll
        for (int m = 1; m < 16; m <<= 1)
            #pragma unroll
            for (int r = 0; r < 8; ++r)
                runmin[t][r] = fminf(runmin[t][r],
                                     __shfl_xor(runmin[t][r], m, 32));
        float sumd = 0.0f;
        #pragma unroll
        for (int r = 0; r < 8; ++r) {
            const int   row = r + (hi ? 8 : 0);
            const float a2r = __shfl(a2[t], row, 32);
            sumd += sqrtf(fmaxf(runmin[t][r] + a2r, 0.0f));
        }
        tot += sumd + __shfl_xor(sumd, 16, 32);
    }
    if (lane == 0) partials[lin] = tot;
}

// ---------------------------------------------------------------------------
// Fallback (no workspace for packing): builds B operands in-loop.
// ---------------------------------------------------------------------------
__global__ __launch_bounds__(32)
void chamfer_rowmin_wmma(const float* __restrict__ pt_src,
                         const float* __restrict__ pt_ref,
                         const float* __restrict__ points_src,
                         const float* __restrict__ points_ref,
                         float* __restrict__ partials)
{
    const int lin   = blockIdx.x;
    const int itile = lin & (NTILES - 1);
    const int b     = (lin >> 8) & (BATCH - 1);
    const int pair  = lin >> 10;

    const float* A; const float* Bp;
    switch (pair) {
        case 0:  A = pt_src; Bp = pt_ref;     break;
        case 1:  A = pt_ref; Bp = pt_src;     break;
        case 2:  A = pt_src; Bp = points_src; break;
        default: A = pt_ref; Bp = points_ref; break;
    }
    A  += (size_t)b * NPTS * 3;
    Bp += (size_t)b * NPTS * 3;

    const int  lane = threadIdx.x;
    const bool hi   = lane >= 16;
    const int  l16  = lane & 15;

    const float* ap = A + (size_t)(itile * TILE + l16) * 3;
    const float ax = ap[0], ay = ap[1], az = ap[2];
    const float a2 = ax * ax + ay * ay + az * az;
    v2f Aop; Aop.x = hi ? az : ax; Aop.y = hi ? 1.0f : ay;

    v8f runmin;
    #pragma unroll
    for (int r = 0; r < 8; ++r) runmin[r] = 3.0e37f;

    #pragma unroll 4
    for (int j = 0; j < NTILES; ++j) {
        const float* bp = Bp + (size_t)(j * TILE + l16) * 3;
        const float bx = bp[0], by = bp[1], bz = bp[2];
        const float b2 = bx * bx + by * by + bz * bz;
        v2f Bop;
        Bop.x = hi ? (-2.0f * bz) : (-2.0f * bx);
        Bop.y = hi ? b2           : (-2.0f * by);
        v8f c = {};
        v8f d = __builtin_amdgcn_wmma_f32_16x16x4_f32(
            false, Aop, false, Bop, (short)0, c, false, false);
        #pragma unroll
        for (int r = 0; r < 8; ++r) runmin[r] = fminf(runmin[r], d[r]);
    }

    #pragma unroll
    for (int m = 1; m < 16; m <<= 1)
        #pragma unroll
        for (int r = 0; r < 8; ++r)
            runmin[r] = fminf(runmin[r], __shfl_xor(runmin[r], m, 32));

    float sumd = 0.0f;
    #pragma unroll
    for (int r = 0; r < 8; ++r) {
        const int   row = r + (hi ? 8 : 0);
        const float a2r = __shfl(a2, row, 32);
        sumd += sqrtf(fmaxf(runmin[r] + a2r, 0.0f));
    }
    const float tot = sumd + __shfl_xor(sumd, 16, 32);
    if (lane == 0) partials[lin] = tot;
}

// Deterministic single-block tree reduction of per-wave partials.
__global__ __launch_bounds__(256)
void chamfer_reduce(const float* __restrict__ partials, int n,
                    float* __restrict__ out, float scale)
{
    __shared__ float s[256];
    float acc = 0.0f;
    for (int i = threadIdx.x; i < n; i += 256) acc += partials[i];
    s[threadIdx.x] = acc;
    __syncthreads();
    for (int k = 128; k > 0; k >>= 1) {
        if (threadIdx.x < k) s[threadIdx.x] += s[threadIdx.x + k];
        __syncthreads();
    }
    if (threadIdx.x == 0) out[0] = s[0] * scale;
}

extern "C" void kernel_launch(void* const* d_in, const int* in_sizes, int n_in,
                              void* d_out, int out_size, void* d_ws, size_t ws_size,
                              hipStream_t stream)
{
    (void)in_sizes; (void)n_in; (void)out_size;
    const float* pt_src     = (const float*)d_in[0];
    const float* pt_ref     = (const float*)d_in[1];
    const float* points_src = (const float*)d_in[2];
    const float* points_ref = (const float*)d_in[3];
    float* out = (float*)d_out;

    const float  scale      = 1.0f / (float)(BATCH * NPTS);
    const size_t pack_bytes = PACK_V2F * sizeof(v2f);        // 1 MB

    if (ws_size >= pack_bytes + (size_t)NBLK_MN * sizeof(float)) {
        v2f*   packed   = (v2f*)d_ws;
        float* partials = (float*)((char*)d_ws + pack_bytes);

        pack_b_operands<<<NBLK_PK, 256, 0, stream>>>(
            pt_src, pt_ref, points_src, points_ref, packed);
        chamfer_rowmin_wmma_packed<<<NBLK_MN, 32, 0, stream>>>(
            pt_src, pt_ref, packed, partials);
        chamfer_reduce<<<1, 256, 0, stream>>>(partials, NBLK_MN, out, scale);
    } else {
        float* partials = (float*)d_ws;                      // 16 KB
        chamfer_rowmin_wmma<<<NBLK_FB, 32, 0, stream>>>(
            pt_src, pt_ref, points_src, points_ref, partials);
        chamfer_reduce<<<1, 256, 0, stream>>>(partials, NBLK_FB, out, scale);
    }
}